// TextLSTMEncoder_43834436223392
// MI455X (gfx1250) — compile-verified
//
#include <hip/hip_runtime.h>

typedef __attribute__((ext_vector_type(2))) float v2f;
typedef __attribute__((ext_vector_type(8))) float v8f;

#define HID   1024
#define EMB   512
#define BATCH 64
#define SEQ   512
#define KWAVES 6          // 2 waves over EMB (2*256) + 4 waves over HID (4*256)
#define KCHUNK 256        // K elements per wave

__device__ __forceinline__ float sigmoidf_(float x) {
    return 1.0f / (1.0f + __expf(-x));
}

// ---------------------------------------------------------------------------
// Zero-init kernel for h/c state in workspace (ws is poisoned 0xAA by harness)
// ---------------------------------------------------------------------------
__global__ void lstm_init_state(float* __restrict__ p, int n) {
    int i = blockIdx.x * blockDim.x + threadIdx.x;
    if (i < n) p[i] = 0.0f;
}

// ---------------------------------------------------------------------------
// Fused LSTM step with 6-way K-split.
//   grid = (HID/16 = 64, BATCH/16 = 4), block = 192 (6 waves)
// Each block owns one 16(batch) x 16(H) output tile. Wave w accumulates the
// four gate tiles (i,f,g,o) over a 256-wide K chunk with
// V_WMMA_F32_16X16X4_F32:
//   waves 0-1 : K in embedding GEMM (A gathered from emb, B = W_ih)
//   waves 2-5 : K in hidden GEMM    (A = h_prev,          B = W_hh)
// Partials are reduced through LDS; wave 0 applies biases + nonlinearities
// and updates c/h.
// ---------------------------------------------------------------------------
__global__ __launch_bounds__(192)
void lstm_step(const int*   __restrict__ ids,     // x [B, T]
               const float* __restrict__ emb,     // [VOCAB, EMB]
               const float* __restrict__ W_ih,    // [4H, E] row-major
               const float* __restrict__ b_ih,    // [4H]
               const float* __restrict__ W_hh,    // [4H, H] row-major
               const float* __restrict__ b_hh,    // [4H]
               const float* __restrict__ h_prev,  // [B, H]
               float*       __restrict__ h_next,  // [B, H]
               float*       __restrict__ c_state, // [B, H]
               int t)
{
    __shared__ float red[KWAVES * 4 * 256];   // 24 KB: [wave][gate][r*32+lane]

    const int tid  = threadIdx.x;
    const int w    = tid >> 5;        // wave 0..5 (K-chunk id)
    const int lane = tid & 31;
    const int nb   = blockIdx.x;      // 0..63 : column block within H
    const int mb   = blockIdx.y;      // 0..3  : batch block
    const int half = lane >> 4;       // K-pair select in A/B fragment
    const int l16  = lane & 15;
    const int m    = mb * 16 + l16;   // A row (batch index) for this lane
    const int n    = nb * 16 + l16;   // B/D column (H index) for this lane

    // Select this wave's K chunk: A row pointer + 4 gate weight row pointers.
    const float* arow;
    const float* brow[4];
    if (w < 2) {
        const int koff = w * KCHUNK;                         // 0 or 256 in EMB
        const int id   = ids[(size_t)m * SEQ + t];
        arow = emb + (size_t)id * EMB + koff;
#pragma unroll
        for (int g = 0; g < 4; ++g)
            brow[g] = W_ih + (size_t)(g * HID + n) * EMB + koff;
    } else {
        const int koff = (w - 2) * KCHUNK;                   // 0..768 in HID
        arow = h_prev + (size_t)m * HID + koff;
#pragma unroll
        for (int g = 0; g < 4; ++g)
            brow[g] = W_hh + (size_t)(g * HID + n) * HID + koff;
    }

    v8f acc[4];
#pragma unroll
    for (int g = 0; g < 4; ++g) acc[g] = v8f{0.f, 0.f, 0.f, 0.f, 0.f, 0.f, 0.f, 0.f};

    // ---- 256-wide K chunk: 64 k-steps x 4 gate WMMAs ----
    for (int k0 = 0; k0 < KCHUNK; k0 += 4) {
        const int kk = k0 + 2 * half;
        v2f a = *(const v2f*)(arow + kk);
#pragma unroll
        for (int g = 0; g < 4; ++g) {
            v2f b = *(const v2f*)(brow[g] + kk);
            acc[g] = __builtin_amdgcn_wmma_f32_16x16x4_f32(
                false, a, false, b, (short)0, acc[g], false, false);
        }
    }

    // ---- cross-wave reduction through LDS ----
    if (w != 0) {
#pragma unroll
        for (int g = 0; g < 4; ++g)
#pragma unroll
            for (int r = 0; r < 8; ++r)
                red[((w * 4 + g) << 8) + r * 32 + lane] = acc[g][r];
    }
    __syncthreads();

    if (w == 0) {
#pragma unroll
        for (int g = 0; g < 4; ++g)
#pragma unroll
            for (int r = 0; r < 8; ++r) {
                float s = acc[g][r];
#pragma unroll
                for (int ww = 1; ww < KWAVES; ++ww)
                    s += red[((ww * 4 + g) << 8) + r * 32 + lane];
                acc[g][r] = s;
            }

        // Per-lane gate biases (column n fixed per lane in the C/D layout)
        float bias[4];
#pragma unroll
        for (int g = 0; g < 4; ++g) bias[g] = b_ih[g * HID + n] + b_hh[g * HID + n];

        // Elementwise LSTM update.
        // C/D layout: VGPR r, lane -> row M = r + 8*(lane/16), col N = lane%16.
#pragma unroll
        for (int r = 0; r < 8; ++r) {
            const int    row = mb * 16 + r + 8 * half;
            const size_t idx = (size_t)row * HID + n;
            float gi = sigmoidf_(acc[0][r] + bias[0]);
            float gf = sigmoidf_(acc[1][r] + bias[1]);
            float gg = tanhf    (acc[2][r] + bias[2]);
            float go = sigmoidf_(acc[3][r] + bias[3]);
            float c_new = gf * c_state[idx] + gi * gg;
            c_state[idx] = c_new;
            h_next[idx]  = go * tanhf(c_new);
        }
    }
}

// ---------------------------------------------------------------------------
// Final FC: out[b][j] = sum_k h_last[b][k] * fc_w[j][k] + fc_b[j]
//   grid = (HID/16 = 64, BATCH/16 = 4), block = 32 (runs once; not critical)
// ---------------------------------------------------------------------------
__global__ __launch_bounds__(32)
void lstm_fc(const float* __restrict__ h_last, // [B, H]
             const float* __restrict__ fc_w,   // [H, H] row-major
             const float* __restrict__ fc_b,   // [H]
             float*       __restrict__ out)    // [B, H]
{
    const int lane = threadIdx.x;
    const int nb   = blockIdx.x;
    const int mb   = blockIdx.y;
    const int half = lane >> 4;
    const int l16  = lane & 15;
    const int m    = mb * 16 + l16;
    const int n    = nb * 16 + l16;

    const float* arow = h_last + (size_t)m * HID;
    const float* brow = fc_w   + (size_t)n * HID;

    v8f acc = v8f{0.f, 0.f, 0.f, 0.f, 0.f, 0.f, 0.f, 0.f};
    for (int k0 = 0; k0 < HID; k0 += 4) {
        const int kk = k0 + 2 * half;
        v2f a = *(const v2f*)(arow + kk);
        v2f b = *(const v2f*)(brow + kk);
        acc = __builtin_amdgcn_wmma_f32_16x16x4_f32(
            false, a, false, b, (short)0, acc, false, false);
    }

    const float bias = fc_b[n];
#pragma unroll
    for (int r = 0; r < 8; ++r) {
        const int row = mb * 16 + r + 8 * half;
        out[(size_t)row * HID + n] = acc[r] + bias;
    }
}

// ---------------------------------------------------------------------------
extern "C" void kernel_launch(void* const* d_in, const int* in_sizes, int n_in,
                              void* d_out, int out_size, void* d_ws, size_t ws_size,
                              hipStream_t stream) {
    const int*   x     = (const int*)  d_in[0];
    const float* emb   = (const float*)d_in[1];
    const float* W_ih  = (const float*)d_in[2];
    const float* b_ih  = (const float*)d_in[3];
    const float* W_hh  = (const float*)d_in[4];
    const float* b_hh  = (const float*)d_in[5];
    const float* fc_w  = (const float*)d_in[6];
    const float* fc_b  = (const float*)d_in[7];
    float*       out   = (float*)d_out;

    // Workspace: h double-buffer + c state (3 * 64*1024 floats = 768 KB)
    float* h0 = (float*)d_ws;
    float* h1 = h0 + BATCH * HID;
    float* c  = h1 + BATCH * HID;

    const int state_elems = 3 * BATCH * HID;
    lstm_init_state<<<(state_elems + 255) / 256, 256, 0, stream>>>(h0, state_elems);

    dim3 grid(HID / 16, BATCH / 16);  // (64, 4)
    for (int t = 0; t < SEQ; ++t) {
        const float* hp = (t & 1) ? h1 : h0;
        float*       hn = (t & 1) ? h0 : h1;
        lstm_step<<<grid, 192, 0, stream>>>(x, emb, W_ih, b_ih, W_hh, b_hh,
                                            hp, hn, c, t);
    }
    // SEQ = 512 (even): last step (t = 511, odd) wrote h0.
    const float* h_last = h0;

    lstm_fc<<<grid, 32, 0, stream>>>(h_last, fc_w, fc_b, out);
}